// TMRPCEN_11467562680726
// MI455X (gfx1250) — compile-verified
//
#include <hip/hip_runtime.h>

// PCEN: EMA scan on the CDNA5 matrix core + streaming pointwise stage.
//
// Roofline (MI455X): out = 16*10*128*4000 f32 = 327.7MB nontemporal stores,
// x = 32.8MB (kept L2-resident, 192MB L2). ~360MB HBM @ 23.3TB/s => ~15.5us.
// Compute floor = 4 raw v_log/v_exp per element (~330M TRANS) — balanced.
//
// Scan blocking: for one (rate n, feature f), a 16-step chunk across all 16
// batches is M(16x16) = U'(16x16) @ X(16x16) + a^(i+1) (x) m_prev, with
// U'[i,j] = s*a^(i-j) lower-triangular (wave-constant) -> 4 chained
// V_WMMA_F32_16X16X4_F32 (K=4). The cross-chunk carry is NOT taken from the
// WMMA result: m[t0+15] = a^16*carry + dot(w, x[t0..t0+15]) is computed
// per-lane, so the serial dependency is a single v_fma per 16 timesteps and
// the WMMA->pcen->store path is pure throughput.
// T is split into 10 segments (dot-only carry fast-forward): 12800 waves.

#define B_DIM   16
#define F_DIM   128
#define T_DIM   4000
#define NRATES  10
#define NSEG    10
#define SEG_T   (T_DIM / NSEG)   // 400, multiple of 16
#define EPS_F   1e-5f
#define LOG2E   1.4426950408889634f

typedef __attribute__((ext_vector_type(2))) float v2f;
typedef __attribute__((ext_vector_type(4))) float v4f;
typedef __attribute__((ext_vector_type(8))) float v8f;

// raw v_exp_f32 / v_log_f32 (base-2); inputs bounded away from denorms here
__device__ __forceinline__ float rexp2(float x) { return __builtin_amdgcn_exp2f(x); }
__device__ __forceinline__ float rlog2(float x) { return __builtin_amdgcn_logf(x); }
__device__ __forceinline__ float fpow(float x, float y) { return rexp2(y * rlog2(x)); }

__device__ __forceinline__ float dot8(const float* w, const v4f a, const v4f b) {
  float p0 = __builtin_fmaf(w[0], a[0], w[1] * a[1]);
  float p1 = __builtin_fmaf(w[2], a[2], w[3] * a[3]);
  float p2 = __builtin_fmaf(w[4], b[0], w[5] * b[1]);
  float p3 = __builtin_fmaf(w[6], b[2], w[7] * b[3]);
  return (p0 + p1) + (p2 + p3);
}

__global__ __launch_bounds__(256) void pcen_wmma_scan(
    const float* __restrict__ x, const float* __restrict__ s_log,
    const float* __restrict__ alpha_log, const float* __restrict__ delta_log,
    const float* __restrict__ r_log, float* __restrict__ out) {
  const int gtid = blockIdx.x * blockDim.x + threadIdx.x;
  // wave index is uniform: scalarize so loop bounds & params use SALU/SMEM
  const int w    = __builtin_amdgcn_readfirstlane(gtid >> 5);
  const int lane = threadIdx.x & 31;
  const int hi   = lane >> 4;            // lane half (K/M split in WMMA layouts)
  const int col  = lane & 15;            // A: row ti ; B/D: column b (batch)

  const int seg = w / (NRATES * F_DIM);
  const int rem = w - seg * (NRATES * F_DIM);
  const int n   = rem >> 7;              // rate
  const int f   = rem & 127;             // feature

  // ---- wave-uniform parameters ----
  const float s     = rexp2(s_log[n * F_DIM + f] * LOG2E);
  const float a     = 1.0f - s;
  const float alpha = rexp2(alpha_log[f] * LOG2E);
  const float delta = rexp2(delta_log[f] * LOG2E);
  const float r     = rexp2(r_log[f] * LOG2E);
  const float dr    = fpow(delta, r);
  const float a16   = fpow(a, 16.0f);

  // ---- A-matrix fragments: U'[ti,tj] = (tj<=ti) ? s*a^(ti-tj) : 0 ----
  // 32-bit 16x4 A layout: lane holds row M=col, K = vgpr + 2*hi.
  v2f u0, u1, u2, u3;
#pragma unroll
  for (int v = 0; v < 2; ++v) {
    int t0j = 0  + v + 2 * hi;
    int t1j = 4  + v + 2 * hi;
    int t2j = 8  + v + 2 * hi;
    int t3j = 12 + v + 2 * hi;
    u0[v] = (t0j <= col) ? s * fpow(a, (float)(col - t0j)) : 0.0f;
    u1[v] = (t1j <= col) ? s * fpow(a, (float)(col - t1j)) : 0.0f;
    u2[v] = (t2j <= col) ? s * fpow(a, (float)(col - t2j)) : 0.0f;
    u3[v] = (t3j <= col) ? s * fpow(a, (float)(col - t3j)) : 0.0f;
  }
  // Carry injection: C[ti,b] = carry_b * a^(ti+1), ti = j + 8*hi
  v8f pj;
#pragma unroll
  for (int j = 0; j < 8; ++j)
    pj[j] = fpow(a, (float)(j + 8 * hi + 1));
  // Carry dot weights: w_t = s*a^(15-t); own half covers t = 8*hi + j,
  // other half covers t = 8*(1-hi) + j.
  float wown[8], woth[8];
#pragma unroll
  for (int j = 0; j < 8; ++j) {
    wown[j] = s * fpow(a, (float)(15 - (8 * hi + j)));
    woth[j] = s * fpow(a, (float)(15 - (8 * (1 - hi) + j)));
  }

  // Per-lane row pointers (lane's batch is b = col)
  const float* xrow = x + ((size_t)col * F_DIM + f) * T_DIM;
  float* orow = out + (((size_t)col * NRATES + n) * F_DIM + f) * T_DIM;

  // m[-1] := x[0]; a+s==1 makes the first step reproduce m[0]=x[0] exactly.
  float carry = xrow[0];

  const int t_start = seg * SEG_T;
  const int t_end   = t_start + SEG_T;

  // ---- Prefix: dot-only carry fast-forward (1 serial FMA / 16 steps) ----
  for (int t0 = 0; t0 < t_start; t0 += 16) {
    v4f xl  = *(const v4f*)(xrow + t0 + 8 * hi);
    v4f xh  = *(const v4f*)(xrow + t0 + 8 * hi + 4);
    v4f xo0 = *(const v4f*)(xrow + t0 + 8 * (1 - hi));
    v4f xo1 = *(const v4f*)(xrow + t0 + 8 * (1 - hi) + 4);
    float d = dot8(wown, xl, xh) + dot8(woth, xo0, xo1);
    carry = __builtin_fmaf(a16, carry, d);
  }

  // ---- Main: WMMA scan + PCEN + nontemporal 16B stores ----
  for (int t0 = t_start; t0 < t_end; t0 += 16) {
    // B fragments X[tj,b]: lane holds tj = 4*kk + 2*hi + {0,1}
    v2f bf0 = *(const v2f*)(xrow + t0 + 0  + 2 * hi);
    v2f bf1 = *(const v2f*)(xrow + t0 + 4  + 2 * hi);
    v2f bf2 = *(const v2f*)(xrow + t0 + 8  + 2 * hi);
    v2f bf3 = *(const v2f*)(xrow + t0 + 12 + 2 * hi);
    // own-half x (D layout, also pcen input): t = t0 + 8*hi + j
    v4f xl  = *(const v4f*)(xrow + t0 + 8 * hi);
    v4f xh  = *(const v4f*)(xrow + t0 + 8 * hi + 4);
    // other-half x, only for the carry dot
    v4f xo0 = *(const v4f*)(xrow + t0 + 8 * (1 - hi));
    v4f xo1 = *(const v4f*)(xrow + t0 + 8 * (1 - hi) + 4);

    if (t0 + 16 < t_end)
      __builtin_prefetch(xrow + t0 + 16, 0, 1);   // global_prefetch_b8

    v8f acc;
#pragma unroll
    for (int j = 0; j < 8; ++j) acc[j] = carry * pj[j];
    acc = __builtin_amdgcn_wmma_f32_16x16x4_f32(false, u0, false, bf0, (short)0, acc, false, false);
    acc = __builtin_amdgcn_wmma_f32_16x16x4_f32(false, u1, false, bf1, (short)0, acc, false, false);
    acc = __builtin_amdgcn_wmma_f32_16x16x4_f32(false, u2, false, bf2, (short)0, acc, false, false);
    acc = __builtin_amdgcn_wmma_f32_16x16x4_f32(false, u3, false, bf3, (short)0, acc, false, false);

    // serial cross-chunk dependency: one FMA (independent of WMMA result)
    float d = dot8(wown, xl, xh) + dot8(woth, xo0, xo1);
    carry = __builtin_fmaf(a16, carry, d);

    // pcen = (x*(eps+m)^(-alpha) + delta)^r - delta^r   (raw v_log/v_exp)
    v4f ylo, yhi;
#pragma unroll
    for (int j = 0; j < 4; ++j) {
      float sm = rexp2(-alpha * rlog2(EPS_F + acc[j]));
      ylo[j] = rexp2(r * rlog2(__builtin_fmaf(xl[j], sm, delta))) - dr;
    }
#pragma unroll
    for (int j = 0; j < 4; ++j) {
      float sm = rexp2(-alpha * rlog2(EPS_F + acc[4 + j]));
      yhi[j] = rexp2(r * rlog2(__builtin_fmaf(xh[j], sm, delta))) - dr;
    }
    // stream past L2: out is never re-read, keep x resident instead
    __builtin_nontemporal_store(ylo, (v4f*)(orow + t0 + 8 * hi));
    __builtin_nontemporal_store(yhi, (v4f*)(orow + t0 + 8 * hi + 4));
  }
}

extern "C" void kernel_launch(void* const* d_in, const int* in_sizes, int n_in,
                              void* d_out, int out_size, void* d_ws, size_t ws_size,
                              hipStream_t stream) {
  const float* x         = (const float*)d_in[0];
  const float* s_log     = (const float*)d_in[1];
  const float* alpha_log = (const float*)d_in[2];
  const float* delta_log = (const float*)d_in[3];
  const float* r_log     = (const float*)d_in[4];
  float* out = (float*)d_out;

  const int waves = NSEG * NRATES * F_DIM;   // 12800 waves, one per (seg,n,f)
  const int block = 256;                     // 8 wave32 waves
  const int grid  = (waves * 32) / block;    // 1600 blocks
  pcen_wmma_scan<<<grid, block, 0, stream>>>(x, s_log, alpha_log, delta_log,
                                             r_log, out);
}